// Block_41205916238091
// MI455X (gfx1250) — compile-verified
//
#include <hip/hip_runtime.h>
#include <math.h>

// ---------------------------------------------------------------------------
// CDNA5 (gfx1250) fp32 WMMA helpers.  V_WMMA_F32_16X16X4_F32:
//   A: 16x4 f32  -> 2 VGPR (lanes 0-15: M=lane, K={0,1}; lanes 16-31: K={2,3})
//   B: 4x16 f32  -> 2 VGPR (lanes 0-15: N=lane, K={0,1}; lanes 16-31: K={2,3})
//   C/D: 16x16   -> 8 VGPR (vgpr i: lanes0-15 M=i, lanes16-31 M=i+8; N=lane&15)
// ---------------------------------------------------------------------------
typedef __attribute__((ext_vector_type(2))) float v2f;
typedef __attribute__((ext_vector_type(8))) float v8f;

__device__ __forceinline__ v8f wmma4(v2f a, v2f b, v8f c) {
  return __builtin_amdgcn_wmma_f32_16x16x4_f32(false, a, false, b, (short)0, c,
                                               false, false);
}

__device__ __forceinline__ float gelu_exact(float x) {
  return 0.5f * x * (1.0f + erff(x * 0.70710678118654752440f));
}

// low 32 bits of a flat shared pointer == LDS byte offset (ISA 10.2)
__device__ __forceinline__ unsigned lds_off(const void* p) {
  return (unsigned)(uintptr_t)p;
}

// async 16B global -> LDS copy; offset applies to BOTH lds and global address
__device__ __forceinline__ void async_b128(unsigned lds, const float* g) {
  asm volatile("global_load_async_to_lds_b128 %0, %1, off"
               :: "v"(lds), "v"(g) : "memory");
}
__device__ __forceinline__ void async_b128_off16(unsigned lds, const float* g) {
  asm volatile("global_load_async_to_lds_b128 %0, %1, off offset:16"
               :: "v"(lds), "v"(g) : "memory");
}
__device__ __forceinline__ void wait_async0() {
  asm volatile("s_wait_asynccnt 0x0" ::: "memory");
}

// token-in-window -> flat spatial index l (H=W=64, 8 windows per branch)
__device__ __forceinline__ int win_l(int e, int w, int n) {
  if (e == 0) { int r = n >> 3, cl = n & 7; return r * 64 + w * 8 + cl; }  // 64x8 strips
  int rl = n >> 6, c = n & 63; return (w * 8 + rl) * 64 + c;               // 8x64 strips
}

// ---------------------------------------------------------------------------
// x (B,C,H,W) -> t (B*HW, C)   (32x32 LDS tile transpose, coalesced both sides)
// ---------------------------------------------------------------------------
__global__ __launch_bounds__(256) void k_tin(const float* __restrict__ x,
                                             float* __restrict__ t) {
  __shared__ float s[32][33];
  const int tx = threadIdx.x & 31, ty = threadIdx.x >> 5;  // 32 x 8
  const int blk = blockIdx.x;
  const int cblk = (blk & 7) * 32;
  const int lblk = ((blk >> 3) & 127) * 32;
  const int b = blk >> 10;
  const float* xb = x + (size_t)b * 256 * 4096;
#pragma unroll
  for (int i = 0; i < 4; ++i)
    s[ty + 8 * i][tx] = xb[(size_t)(cblk + ty + 8 * i) * 4096 + lblk + tx];
  __syncthreads();
  float* tb = t + (size_t)b * 4096 * 256;
#pragma unroll
  for (int i = 0; i < 4; ++i)
    tb[(size_t)(lblk + ty + 8 * i) * 256 + cblk + tx] = s[tx][ty + 8 * i];
}

// t (B*HW, C) -> out (B,C,H,W)
__global__ __launch_bounds__(256) void k_tout(const float* __restrict__ t,
                                              float* __restrict__ o) {
  __shared__ float s[32][33];
  const int tx = threadIdx.x & 31, ty = threadIdx.x >> 5;
  const int blk = blockIdx.x;
  const int cblk = (blk & 7) * 32;
  const int lblk = ((blk >> 3) & 127) * 32;
  const int b = blk >> 10;
  const float* tb = t + (size_t)b * 4096 * 256;
#pragma unroll
  for (int i = 0; i < 4; ++i)
    s[ty + 8 * i][tx] = tb[(size_t)(lblk + ty + 8 * i) * 256 + cblk + tx];
  __syncthreads();
  float* ob = o + (size_t)b * 256 * 4096;
#pragma unroll
  for (int i = 0; i < 4; ++i)
    ob[(size_t)(cblk + ty + 8 * i) * 4096 + lblk + tx] = s[tx][ty + 8 * i];
}

// per-row mean / rstd over C=256 (LayerNorm statistics)
__global__ __launch_bounds__(256) void k_row_stats(const float* __restrict__ A,
                                                   float* __restrict__ mu,
                                                   float* __restrict__ rstd) {
  __shared__ float red[256];
  const int row = blockIdx.x, c = threadIdx.x;
  const float v = A[(size_t)row * 256 + c];
  red[c] = v;
  __syncthreads();
  for (int off = 128; off > 0; off >>= 1) {
    if (c < off) red[c] += red[c + off];
    __syncthreads();
  }
  const float m = red[0] * (1.0f / 256.0f);
  __syncthreads();
  const float d = v - m;
  red[c] = d * d;
  __syncthreads();
  for (int off = 128; off > 0; off >>= 1) {
    if (c < off) red[c] += red[c + off];
    __syncthreads();
  }
  if (c == 0) {
    mu[row] = m;
    rstd[row] = rsqrtf(red[0] * (1.0f / 256.0f) + 1e-5f);
  }
}

// ---------------------------------------------------------------------------
// Fused GEMM: out = epilogue( LN?(A) @ W^T )
//   A: MxK row-major, W: NxK row-major, out: MxN
// Block tile 128x128x16; 8 waves as 4(m) x 2(n); wave tile 32x64 (2x4 WMMA).
// LDS tiles row-major [row][k] with stride 20 (16B-aligned rows, bank-spread,
// contiguous v2f fragments -> ds_load_b64, contiguous 16B for async copies).
// No-LN path stages tiles with global_load_async_to_lds_b128 (ASYNCcnt).
// ---------------------------------------------------------------------------
__global__ __launch_bounds__(256) void k_gemm(
    const float* __restrict__ A, const float* __restrict__ W,
    const float* __restrict__ bias, const float* __restrict__ resid,
    const float* __restrict__ mu, const float* __restrict__ rstd,
    const float* __restrict__ lnw, const float* __restrict__ lnb,
    float* __restrict__ out, int M, int N, int K, int act) {
  __shared__ float As[128 * 20];
  __shared__ float Bs[128 * 20];
  const int tid = threadIdx.x;
  const int bm = blockIdx.y * 128;
  const int bn = blockIdx.x * 128;
  const int wv = tid >> 5, lane = tid & 31;
  const int lm = lane & 15, kh = lane >> 4;
  const int wm = (wv & 3) * 32, wn = (wv >> 2) * 64;

  const v8f z8 = {0.f, 0.f, 0.f, 0.f, 0.f, 0.f, 0.f, 0.f};
  v8f acc[2][4];
#pragma unroll
  for (int i = 0; i < 2; ++i)
#pragma unroll
    for (int j = 0; j < 4; ++j) acc[i][j] = z8;

  const int srow = tid >> 1;        // 0..127  (row of A / row of W)
  const int skk = (tid & 1) * 8;    // 0 / 8

  float lmu = 0.f, lrs = 1.f;
  if (lnw) { lmu = mu[bm + srow]; lrs = rstd[bm + srow]; }
  const unsigned ldsA = lds_off(&As[srow * 20 + skk]);
  const unsigned ldsB = lds_off(&Bs[srow * 20 + skk]);

  for (int k0 = 0; k0 < K; k0 += 16) {
    __syncthreads();
    const float* ap = A + (size_t)(bm + srow) * K + (k0 + skk);
    const float* wp = W + (size_t)(bn + srow) * K + (k0 + skk);
    if (lnw) {
      // fused-LayerNorm staging (synchronous) + prefetch of next chunk
      float4 a0 = *(const float4*)ap;
      float4 a1 = *(const float4*)(ap + 4);
      a0.x = (a0.x - lmu) * lrs * lnw[k0 + skk + 0] + lnb[k0 + skk + 0];
      a0.y = (a0.y - lmu) * lrs * lnw[k0 + skk + 1] + lnb[k0 + skk + 1];
      a0.z = (a0.z - lmu) * lrs * lnw[k0 + skk + 2] + lnb[k0 + skk + 2];
      a0.w = (a0.w - lmu) * lrs * lnw[k0 + skk + 3] + lnb[k0 + skk + 3];
      a1.x = (a1.x - lmu) * lrs * lnw[k0 + skk + 4] + lnb[k0 + skk + 4];
      a1.y = (a1.y - lmu) * lrs * lnw[k0 + skk + 5] + lnb[k0 + skk + 5];
      a1.z = (a1.z - lmu) * lrs * lnw[k0 + skk + 6] + lnb[k0 + skk + 6];
      a1.w = (a1.w - lmu) * lrs * lnw[k0 + skk + 7] + lnb[k0 + skk + 7];
      *(float4*)&As[srow * 20 + skk] = a0;
      *(float4*)&As[srow * 20 + skk + 4] = a1;
      float4 b0 = *(const float4*)wp;
      float4 b1 = *(const float4*)(wp + 4);
      *(float4*)&Bs[srow * 20 + skk] = b0;
      *(float4*)&Bs[srow * 20 + skk + 4] = b1;
      if (k0 + 16 < K) {
        __builtin_prefetch((const void*)(ap + 16), 0, 3);
        __builtin_prefetch((const void*)(wp + 16), 0, 3);
      }
    } else {
      // raw tiles: async global -> LDS (tracked by ASYNCcnt)
      async_b128(ldsA, ap);
      async_b128_off16(ldsA, ap);
      async_b128(ldsB, wp);
      async_b128_off16(ldsB, wp);
    }
    wait_async0();
    __syncthreads();
#pragma unroll
    for (int s = 0; s < 4; ++s) {
      const int kb = s * 4 + 2 * kh;  // even -> aligned v2f pair
      v2f af[2], bf[4];
#pragma unroll
      for (int i = 0; i < 2; ++i)
        af[i] = *(const v2f*)&As[(wm + i * 16 + lm) * 20 + kb];
#pragma unroll
      for (int j = 0; j < 4; ++j)
        bf[j] = *(const v2f*)&Bs[(wn + j * 16 + lm) * 20 + kb];
#pragma unroll
      for (int i = 0; i < 2; ++i)
#pragma unroll
        for (int j = 0; j < 4; ++j) acc[i][j] = wmma4(af[i], bf[j], acc[i][j]);
    }
  }
  // --- epilogue: +bias, GELU, +residual ---
#pragma unroll
  for (int i = 0; i < 2; ++i) {
#pragma unroll
    for (int j = 0; j < 4; ++j) {
      const int n = bn + wn + j * 16 + lm;
#pragma unroll
      for (int e2 = 0; e2 < 8; ++e2) {
        const int m = bm + wm + i * 16 + e2 + 8 * kh;
        float v = acc[i][j][e2];
        if (bias) v += bias[n];
        if (act == 1) v = gelu_exact(v);
        if (resid) v += resid[(size_t)m * N + n];
        out[(size_t)m * N + n] = v;
      }
    }
  }
}

// ---------------------------------------------------------------------------
// Cross-shaped-window flash attention, both branches.
// Grid: (b, e, window, head) -> 1024 blocks x 256 threads (8 waves).
// Wave owns 16-query tiles; streams 32 key tiles with online softmax.
// QK^T: 2 x wmma (K=8).  P@V: 4 x wmma (K=16, V zero-padded N 8->16).
// All fragments are contiguous aligned pairs -> ds_load_b64.
// LDS: Qs 512x8, Ks 512x10, Vt 8x516 (transposed), Ps 8x16x18  (= 62.6 KB)
// ---------------------------------------------------------------------------
__global__ __launch_bounds__(256) void k_attn(const float* __restrict__ qkv,
                                              float* __restrict__ out) {
  __shared__ float Qs[512 * 8];
  __shared__ float Ks[512 * 10];
  __shared__ float Vt[8 * 516];
  __shared__ float Ps[8 * 16 * 18];
  const int blk = blockIdx.x;
  const int h = blk & 15, w = (blk >> 4) & 7, e = (blk >> 7) & 1, b = blk >> 8;
  const int tid = threadIdx.x;
  const int coff = e * 128 + h * 8;
  const float sc = 0.35355339059327373f;  // 8^-0.5

  for (int n = tid; n < 512; n += 256) {
    const int l = win_l(e, w, n);
    const float* base = qkv + ((size_t)(b * 4096 + l)) * 768 + coff;
    float4 q0 = *(const float4*)(base);
    float4 q1 = *(const float4*)(base + 4);
    float4 kk0 = *(const float4*)(base + 256);
    float4 kk1 = *(const float4*)(base + 260);
    float4 v0 = *(const float4*)(base + 512);
    float4 v1 = *(const float4*)(base + 516);
    float* Q = Qs + n * 8;
    float* K = Ks + n * 10;
    Q[0] = q0.x * sc; Q[1] = q0.y * sc; Q[2] = q0.z * sc; Q[3] = q0.w * sc;
    Q[4] = q1.x * sc; Q[5] = q1.y * sc; Q[6] = q1.z * sc; Q[7] = q1.w * sc;
    K[0] = kk0.x; K[1] = kk0.y; K[2] = kk0.z; K[3] = kk0.w;
    K[4] = kk1.x; K[5] = kk1.y; K[6] = kk1.z; K[7] = kk1.w;
    Vt[0 * 516 + n] = v0.x; Vt[1 * 516 + n] = v0.y;
    Vt[2 * 516 + n] = v0.z; Vt[3 * 516 + n] = v0.w;
    Vt[4 * 516 + n] = v1.x; Vt[5 * 516 + n] = v1.y;
    Vt[6 * 516 + n] = v1.z; Vt[7 * 516 + n] = v1.w;
  }
  __syncthreads();

  const int wv = tid >> 5, lane = tid & 31;
  const int lm = lane & 15, kh = lane >> 4;
  float* Pw = Ps + wv * (16 * 18);
  const v8f z8 = {0.f, 0.f, 0.f, 0.f, 0.f, 0.f, 0.f, 0.f};
  const v2f zp = {0.f, 0.f};

  for (int qt = wv; qt < 32; qt += 8) {
    const v2f qa0 = *(const v2f*)&Qs[(qt * 16 + lm) * 8 + 2 * kh];
    const v2f qa1 = *(const v2f*)&Qs[(qt * 16 + lm) * 8 + 4 + 2 * kh];
    v8f o = z8;
    float mrun[8], lrun[8];
#pragma unroll
    for (int i = 0; i < 8; ++i) { mrun[i] = -3.4e38f; lrun[i] = 0.f; }

    for (int kt = 0; kt < 32; ++kt) {
      // S tile = q @ k^T
      v8f s8 = z8;
      const v2f kb0 = *(const v2f*)&Ks[(kt * 16 + lm) * 10 + 2 * kh];
      const v2f kb1 = *(const v2f*)&Ks[(kt * 16 + lm) * 10 + 4 + 2 * kh];
      s8 = wmma4(qa0, kb0, s8);
      s8 = wmma4(qa1, kb1, s8);
      // online softmax (row r: vgpr r%8, lane-half r/8; reduce over 16 lanes)
      float nm[8], al[8];
      v8f p8;
#pragma unroll
      for (int i = 0; i < 8; ++i) {
        float r = s8[i];
        r = fmaxf(r, __shfl_xor(r, 8, 16));
        r = fmaxf(r, __shfl_xor(r, 4, 16));
        r = fmaxf(r, __shfl_xor(r, 2, 16));
        r = fmaxf(r, __shfl_xor(r, 1, 16));
        nm[i] = fmaxf(mrun[i], r);
        al[i] = __expf(mrun[i] - nm[i]);
        mrun[i] = nm[i];
      }
#pragma unroll
      for (int i = 0; i < 8; ++i) {
        float pv = __expf(s8[i] - nm[i]);
        p8[i] = pv;
        float r = pv;
        r += __shfl_xor(r, 8, 16);
        r += __shfl_xor(r, 4, 16);
        r += __shfl_xor(r, 2, 16);
        r += __shfl_xor(r, 1, 16);
        lrun[i] = lrun[i] * al[i] + r;
        o[i] *= al[i];
      }
      // re-layout P (C-layout) -> A-layout via per-wave LDS scratch
#pragma unroll
      for (int i = 0; i < 8; ++i) Pw[(i + 8 * kh) * 18 + lm] = p8[i];
      __builtin_amdgcn_wave_barrier();
      // O += P @ V
#pragma unroll
      for (int s = 0; s < 4; ++s) {
        const int kk = 4 * s + 2 * kh;
        const v2f pa = *(const v2f*)&Pw[lm * 18 + kk];
        v2f vb = zp;
        if (lm < 8) vb = *(const v2f*)&Vt[lm * 516 + kt * 16 + kk];
        o = wmma4(pa, vb, o);
      }
      __builtin_amdgcn_wave_barrier();
    }
    // write normalized rows (only d = lm < 8 are real)
    if (lm < 8) {
#pragma unroll
      for (int i = 0; i < 8; ++i) {
        const int n = qt * 16 + i + 8 * kh;
        const int l = win_l(e, w, n);
        out[((size_t)(b * 4096 + l)) * 256 + coff + lm] = o[i] / lrun[i];
      }
    }
  }
}

// ---------------------------------------------------------------------------
// LePE: depthwise 3x3 'SAME' conv on each window image of V, += into attn out.
// ---------------------------------------------------------------------------
__global__ __launch_bounds__(256) void k_lepe(
    const float* __restrict__ qkv, const float* __restrict__ w0,
    const float* __restrict__ b0, const float* __restrict__ w1,
    const float* __restrict__ b1, float* __restrict__ out) {
  const int idx = blockIdx.x * 256 + threadIdx.x;
  const int c = idx & 127;
  const int l = (idx >> 7) & 4095;
  const int b = (idx >> 19) & 3;
  const int e = idx >> 21;
  const float* cw = (e == 0) ? w0 : w1;
  const float* cb = (e == 0) ? b0 : b1;
  const int r = l >> 6, col = l & 63;
  float sum = cb[c];
  const float* vbase = qkv + (size_t)b * 4096 * 768 + 512 + e * 128 + c;
#pragma unroll
  for (int dr = -1; dr <= 1; ++dr) {
#pragma unroll
    for (int dc = -1; dc <= 1; ++dc) {
      const int rr = r + dr, cc = col + dc;
      bool ok;
      if (e == 0) ok = (rr >= 0) && (rr < 64) && ((cc >> 3) == (col >> 3));
      else        ok = (cc >= 0) && (cc < 64) && ((rr >> 3) == (r >> 3));
      if (ok)
        sum += vbase[(size_t)(rr * 64 + cc) * 768] *
               cw[c * 9 + (dr + 1) * 3 + (dc + 1)];
    }
  }
  out[((size_t)(b * 4096 + l)) * 256 + e * 128 + c] += sum;
}

// ---------------------------------------------------------------------------
extern "C" void kernel_launch(void* const* d_in, const int* in_sizes, int n_in,
                              void* d_out, int out_size, void* d_ws,
                              size_t ws_size, hipStream_t stream) {
  (void)in_sizes; (void)n_in; (void)out_size; (void)ws_size;
  const float* x      = (const float*)d_in[0];
  const float* ln1_w  = (const float*)d_in[1];
  const float* ln1_b  = (const float*)d_in[2];
  const float* qkv_w  = (const float*)d_in[3];
  const float* conv0w = (const float*)d_in[4];
  const float* conv0b = (const float*)d_in[5];
  const float* conv1w = (const float*)d_in[6];
  const float* conv1b = (const float*)d_in[7];
  const float* proj_w = (const float*)d_in[8];
  const float* proj_b = (const float*)d_in[9];
  const float* ln2_w  = (const float*)d_in[10];
  const float* ln2_b  = (const float*)d_in[11];
  const float* mlp_w1 = (const float*)d_in[12];
  const float* mlp_b1 = (const float*)d_in[13];
  const float* mlp_w2 = (const float*)d_in[14];
  const float* mlp_b2 = (const float*)d_in[15];
  float* outp = (float*)d_out;

  const size_t Mr = 16384;  // B * H * W
  float* ws     = (float*)d_ws;
  float* t_buf  = ws;                // 16384 x 256   (t; later final out)
  float* ab_buf = t_buf + Mr * 256;  // 16384 x 256   (attn out)
  float* big    = ab_buf + Mr * 256; // 16384 x 1024  (qkv 768-wide; later h1)
  float* t2_buf = big + Mr * 1024;   // 16384 x 256
  float* mu     = t2_buf + Mr * 256; // 16384
  float* rstd   = mu + Mr;           // 16384

  // x -> t, LN1 stats
  k_tin<<<4096, 256, 0, stream>>>(x, t_buf);
  k_row_stats<<<16384, 256, 0, stream>>>(t_buf, mu, rstd);
  // qkv = LN1(t) @ qkv_w^T        (16384 x 768 x 256)
  k_gemm<<<dim3(6, 128), 256, 0, stream>>>(t_buf, qkv_w, nullptr, nullptr, mu,
                                           rstd, ln1_w, ln1_b, big, 16384, 768,
                                           256, 0);
  // cross-shaped window flash attention (both branches) + LePE
  k_attn<<<1024, 256, 0, stream>>>(big, ab_buf);
  k_lepe<<<16384, 256, 0, stream>>>(big, conv0w, conv0b, conv1w, conv1b,
                                    ab_buf);
  // t2 = t + attn @ proj_w^T + proj_b          (async-LDS staging path)
  k_gemm<<<dim3(2, 128), 256, 0, stream>>>(ab_buf, proj_w, proj_b, t_buf,
                                           nullptr, nullptr, nullptr, nullptr,
                                           t2_buf, 16384, 256, 256, 0);
  // LN2 stats; h1 = gelu(LN2(t2) @ mlp_w1^T + b1)   (reuses qkv buffer)
  k_row_stats<<<16384, 256, 0, stream>>>(t2_buf, mu, rstd);
  k_gemm<<<dim3(8, 128), 256, 0, stream>>>(t2_buf, mlp_w1, mlp_b1, nullptr, mu,
                                           rstd, ln2_w, ln2_b, big, 16384, 1024,
                                           256, 1);
  // out = t2 + h1 @ mlp_w2^T + b2   (async-LDS staging; into t_buf, now dead)
  k_gemm<<<dim3(2, 128), 256, 0, stream>>>(big, mlp_w2, mlp_b2, t2_buf, nullptr,
                                           nullptr, nullptr, nullptr, t_buf,
                                           16384, 256, 1024, 0);
  // t -> (B, C, H, W)
  k_tout<<<4096, 256, 0, stream>>>(t_buf, outp);
}